// MutualInformationLoss_4466765988329
// MI455X (gfx1250) — compile-verified
//
#include <hip/hip_runtime.h>
#include <math.h>

#define NBINS 256
#define NCELLS (NBINS * NBINS)
#define EPSF 1e-8f

typedef __attribute__((ext_vector_type(2))) float v2f;
typedef __attribute__((ext_vector_type(4))) float f4;
typedef __attribute__((ext_vector_type(8))) float v8f;

// ---- monotone uint encoding of float for atomic min/max (handles sign) ----
__device__ __forceinline__ unsigned enc_f32(float f) {
    unsigned u = __float_as_uint(f);
    return (u & 0x80000000u) ? ~u : (u | 0x80000000u);
}
__device__ __forceinline__ float dec_f32(unsigned e) {
    unsigned u = (e & 0x80000000u) ? (e & 0x7FFFFFFFu) : ~e;
    return __uint_as_float(u);
}

__device__ __forceinline__ float wave_min(float v) {
#pragma unroll
    for (int o = 16; o > 0; o >>= 1) v = fminf(v, __shfl_xor(v, o, 32));
    return v;
}
__device__ __forceinline__ float wave_max(float v) {
#pragma unroll
    for (int o = 16; o > 0; o >>= 1) v = fmaxf(v, __shfl_xor(v, o, 32));
    return v;
}
__device__ __forceinline__ float wave_sum(float v) {
#pragma unroll
    for (int o = 16; o > 0; o >>= 1) v += __shfl_xor(v, o, 32);
    return v;
}

// ---------------------------------------------------------------------------
// Kernel 0: zero the global histogram, init min/max slots.
// mm[0]=enc(min_x) (atomicMin), mm[1]=enc(max_x) (atomicMax), mm[2..3] = y.
// ---------------------------------------------------------------------------
__global__ void k_init(unsigned* __restrict__ ghist, unsigned* __restrict__ mm) {
    int i = blockIdx.x * blockDim.x + threadIdx.x;
    if (i < NCELLS) ghist[i] = 0u;
    if (i < 4) mm[i] = (i & 1) ? 0u : 0xFFFFFFFFu;
}

// ---------------------------------------------------------------------------
// Kernel 1: fused global min/max of x and y. Streaming b128 NT loads.
// ---------------------------------------------------------------------------
__global__ void k_minmax(const f4* __restrict__ x, const f4* __restrict__ y,
                         unsigned* __restrict__ mm, int n4) {
    const int gid = blockIdx.x * blockDim.x + threadIdx.x;
    const int stride = gridDim.x * blockDim.x;
    float mnx = INFINITY, mxx = -INFINITY, mny = INFINITY, mxy = -INFINITY;
    for (int i = gid; i < n4; i += stride) {
        f4 a = __builtin_nontemporal_load(x + i);
        f4 b = __builtin_nontemporal_load(y + i);
#pragma unroll
        for (int c = 0; c < 4; ++c) {
            mnx = fminf(mnx, a[c]);
            mxx = fmaxf(mxx, a[c]);
            mny = fminf(mny, b[c]);
            mxy = fmaxf(mxy, b[c]);
        }
    }
    mnx = wave_min(mnx);
    mxx = wave_max(mxx);
    mny = wave_min(mny);
    mxy = wave_max(mxy);
    if ((threadIdx.x & 31) == 0) {
        atomicMin(&mm[0], enc_f32(mnx));
        atomicMax(&mm[1], enc_f32(mxx));
        atomicMin(&mm[2], enc_f32(mny));
        atomicMax(&mm[3], enc_f32(mxy));
    }
}

// ---------------------------------------------------------------------------
// Kernel 2: joint histogram. Full 256x256 u32 histogram PRIVATE IN LDS
// (256 KB of the WGP's 320 KB) -> LDS atomics, then one global atomic
// merge per cell per workgroup.
// ---------------------------------------------------------------------------
__global__ void __launch_bounds__(1024, 1)
k_hist(const f4* __restrict__ x, const f4* __restrict__ y,
       const unsigned* __restrict__ mm, unsigned* __restrict__ ghist, int n4) {
    __shared__ unsigned lhist[NCELLS];  // 262144 bytes of LDS
    for (int i = threadIdx.x; i < NCELLS; i += blockDim.x) lhist[i] = 0u;
    __syncthreads();

    const float mnx = dec_f32(mm[0]);
    const float mxx = dec_f32(mm[1]);
    const float mny = dec_f32(mm[2]);
    const float mxy = dec_f32(mm[3]);
    const float sx = (float)NBINS / (mxx - mnx + EPSF);
    const float sy = (float)NBINS / (mxy - mny + EPSF);

    const int gid = blockIdx.x * blockDim.x + threadIdx.x;
    const int stride = gridDim.x * blockDim.x;
    for (int i = gid; i < n4; i += stride) {
        f4 a = __builtin_nontemporal_load(x + i);
        f4 b = __builtin_nontemporal_load(y + i);
#pragma unroll
        for (int c = 0; c < 4; ++c) {
            int xi = (int)floorf((a[c] - mnx) * sx);
            xi = min(max(xi, 0), NBINS - 1);
            int yi = (int)floorf((b[c] - mny) * sy);
            yi = min(max(yi, 0), NBINS - 1);
            atomicAdd(&lhist[(xi << 8) | yi], 1u);
        }
    }
    __syncthreads();

    for (int i = threadIdx.x; i < NCELLS; i += blockDim.x) {
        unsigned v = lhist[i];
        if (v) atomicAdd(&ghist[i], v);
    }
}

// ---------------------------------------------------------------------------
// Kernel 3: finalize. One block, 16 waves.
//  - Marginals via V_WMMA_F32_16X16X4_F32 chains:
//      row sums : D = H_tile(16x4) x Ones(4x16)  (every column of D = rowsum)
//      col sums : D = Ones(16x4)   x H_tile(4x16) (every row of D = colsum)
//  - Then MI + entropies with log-sum reductions.
// ---------------------------------------------------------------------------
__global__ void __launch_bounds__(512) k_final(const unsigned* __restrict__ ghist,
                                               float* __restrict__ out, int n) {
    __shared__ float s_px[NBINS];
    __shared__ float s_py[NBINS];
    __shared__ float s_red[3][16];

    const int tid = threadIdx.x;
    const int wave = tid >> 5;   // 0..15
    const int lane = tid & 31;
    const int half = lane >> 4;  // 0/1
    const int m = lane & 15;

    // ---- marginals (counts) via f32 WMMA ----
    {
        const int base = wave * 16;  // this wave owns rows/cols [base, base+16)
        v8f accR = {};
        v8f accC = {};
        const v2f ones = {1.0f, 1.0f};
        for (int kb = 0; kb < NBINS; kb += 4) {
            // A tile (16x4): lanes 0-15 hold K=kb,kb+1; lanes 16-31 hold K=kb+2,kb+3
            v2f a;
            {
                const int r = base + m;
                const int c = kb + 2 * half;
                a.x = (float)ghist[(r << 8) + c];
                a.y = (float)ghist[(r << 8) + c + 1];
            }
            accR = __builtin_amdgcn_wmma_f32_16x16x4_f32(
                false, a, false, ones, (short)0, accR, false, false);

            // B tile (4x16): lanes 0-15 hold K=kb,kb+1; lanes 16-31 K=kb+2,kb+3
            v2f b;
            {
                const int r = kb + 2 * half;
                const int c = base + m;
                b.x = (float)ghist[(r << 8) + c];
                b.y = (float)ghist[((r + 1) << 8) + c];
            }
            accC = __builtin_amdgcn_wmma_f32_16x16x4_f32(
                false, ones, false, b, (short)0, accC, false, false);
        }
        // D layout: VGPR r, lanes 0-15 -> M=r; lanes 16-31 -> M=8+r
        if (m == 0) {
#pragma unroll
            for (int r = 0; r < 8; ++r) s_px[base + half * 8 + r] = accR[r];
        }
        if (half == 0) s_py[base + m] = accC[0];
    }
    __syncthreads();

    // ---- MI and entropies ----
    const float inv_n = 1.0f / (float)n;
    float mi = 0.0f;
    for (int i = tid; i < NCELLS; i += 512) {
        float cnt = (float)ghist[i];
        if (cnt > 0.0f) {
            float p = cnt * inv_n;
            float pr = s_px[i >> 8] * inv_n;
            float pc = s_py[i & (NBINS - 1)] * inv_n;
            if (pr > 0.0f && pc > 0.0f) mi += p * __logf(p / (pr * pc));
        }
    }
    float hx = 0.0f, hy = 0.0f;
    for (int i = tid; i < NBINS; i += 512) {
        float pr = s_px[i] * inv_n;
        float pc = s_py[i] * inv_n;
        hx -= pr * __logf(pr + EPSF);
        hy -= pc * __logf(pc + EPSF);
    }

    mi = wave_sum(mi);
    hx = wave_sum(hx);
    hy = wave_sum(hy);
    if (lane == 0) {
        s_red[0][wave] = mi;
        s_red[1][wave] = hx;
        s_red[2][wave] = hy;
    }
    __syncthreads();
    if (tid == 0) {
        float M = 0.0f, X = 0.0f, Y = 0.0f;
#pragma unroll
        for (int w = 0; w < 16; ++w) {
            M += s_red[0][w];
            X += s_red[1][w];
            Y += s_red[2][w];
        }
        out[0] = -(M / (sqrtf(X * Y) + EPSF));
    }
}

// ---------------------------------------------------------------------------
extern "C" void kernel_launch(void* const* d_in, const int* in_sizes, int n_in,
                              void* d_out, int out_size, void* d_ws, size_t ws_size,
                              hipStream_t stream) {
    const float* x = (const float*)d_in[0];
    const float* y = (const float*)d_in[1];
    unsigned* ghist = (unsigned*)d_ws;            // 65536 u32 = 256 KB
    unsigned* mm = ghist + NCELLS;                // 4 u32 min/max slots
    const int n = in_sizes[0];                    // 4096*8192 = 33554432
    const int n4 = n / 4;                         // divisible

    k_init<<<(NCELLS + 255) / 256, 256, 0, stream>>>(ghist, mm);
    k_minmax<<<2048, 256, 0, stream>>>((const f4*)x, (const f4*)y, mm, n4);
    k_hist<<<192, 1024, 0, stream>>>((const f4*)x, (const f4*)y, mm, ghist, n4);
    k_final<<<1, 512, 0, stream>>>(ghist, (float*)d_out, n);
}